// GivenPointSliceSamplingSplineWarpSSM_55860344651824
// MI455X (gfx1250) — compile-verified
//
#include <hip/hip_runtime.h>

typedef __attribute__((ext_vector_type(2))) float v2f;
typedef __attribute__((ext_vector_type(8))) float v8f;

#define S_    16
#define SZ_   128
#define NCP_  128
#define NSYS  132          // NCP + 4 (affine)
#define NCOLS (NSYS + 3)   // system + 3 RHS columns
#define LDA   136          // padded row stride for elimination matrix
#define Q_    (S_ * SZ_ * SZ_)

// float offsets inside d_ws
#define P_RM    0     // 9  floats (3x3 rotation, row-major)
#define P_TOFF  16    // 16x4 floats (per-slice total offset xyz, pad)
#define P_BMAT  96    // 128x4 floats (-2tpx,-2tpy,-2tpz,|tp|^2)
#define P_W     640   // 128x4 floats (spline weights w[j][0..2], pad)
#define P_V     1152  // 4x4 floats  (affine v rows, pad)
#define P_ELIM  4096  // 132x136 floats elimination workspace

// ---------------------------------------------------------------------------
// Kernel 0: rotation, offsets, TPS system build + solve (1 block, 256 thr)
// ---------------------------------------------------------------------------
__global__ __launch_bounds__(256) void setup_kernel(
    const float* __restrict__ wcp,    // warped_control_points 128x3
    const float* __restrict__ goff,   // global_offsets 3
    const float* __restrict__ pso,    // per_slice_offsets 16x3
    const float* __restrict__ grot,   // global_rotations 3
    const float* __restrict__ cp,     // control_points 128x3
    const float* __restrict__ coord,  // coordinate_system 16x3x3
    float* __restrict__ ws)
{
  __shared__ float sRm[9];
  __shared__ float sTp[NCP_][3];
  __shared__ int   sPiv;
  const int tid = threadIdx.x;
  float* M = ws + P_ELIM;  // [NSYS][LDA], cols 132..134 = RHS

  // --- rotation matrix Rm = Rz @ Ry @ Rx ---
  if (tid == 0) {
    float th = grot[0], ph = grot[1], ps = grot[2];
    float cth = cosf(th), sth = sinf(th);
    float cph = cosf(ph), sph = sinf(ph);
    float cps = cosf(ps), sps = sinf(ps);
    float Rx[9] = {1, 0, 0, 0, cth, sth, 0, -sth, cth};
    float Ry[9] = {cph, 0, -sph, 0, 1, 0, sph, 0, cph};
    float Rz[9] = {cps, sps, 0, -sps, cps, 0, 0, 0, 1};
    float Rzy[9];
    for (int i = 0; i < 3; ++i)
      for (int j = 0; j < 3; ++j) {
        float a = 0.f;
        for (int k = 0; k < 3; ++k) a += Rz[i*3+k] * Ry[k*3+j];
        Rzy[i*3+j] = a;
      }
    for (int i = 0; i < 3; ++i)
      for (int j = 0; j < 3; ++j) {
        float a = 0.f;
        for (int k = 0; k < 3; ++k) a += Rzy[i*3+k] * Rx[k*3+j];
        sRm[i*3+j] = a;
        ws[P_RM + i*3 + j] = a;
      }
  }
  __syncthreads();

  // --- per-slice total offset: goff + pso[s] @ (coord[s] @ Rm) ---
  if (tid < S_) {
    float rc[9];
    for (int i = 0; i < 3; ++i)
      for (int j = 0; j < 3; ++j) {
        float a = 0.f;
        for (int k = 0; k < 3; ++k) a += coord[(tid*3 + i)*3 + k] * sRm[k*3 + j];
        rc[i*3+j] = a;
      }
    for (int k = 0; k < 3; ++k) {
      float a = 0.f;
      for (int j = 0; j < 3; ++j) a += pso[tid*3 + j] * rc[j*3 + k];
      ws[P_TOFF + tid*4 + k] = a + goff[k];
    }
    ws[P_TOFF + tid*4 + 3] = 0.f;
  }

  // --- tp = flip(wcp)*2-1 ; stage-1 B-matrix columns ---
  if (tid < NCP_) {
    float x = wcp[tid*3 + 2] * 2.f - 1.f;
    float y = wcp[tid*3 + 1] * 2.f - 1.f;
    float z = wcp[tid*3 + 0] * 2.f - 1.f;
    sTp[tid][0] = x; sTp[tid][1] = y; sTp[tid][2] = z;
    ws[P_BMAT + tid*4 + 0] = -2.f * x;
    ws[P_BMAT + tid*4 + 1] = -2.f * y;
    ws[P_BMAT + tid*4 + 2] = -2.f * z;
    ws[P_BMAT + tid*4 + 3] = x*x + y*y + z*z;
  }
  __syncthreads();

  // --- build [A Bm; Bm^T 0 | rhs] system in global scratch ---
  for (int e = tid; e < NSYS * NCOLS; e += 256) {
    int i = e / NCOLS;
    int j = e % NCOLS;
    float val;
    if (j < NSYS) {
      if (i < NCP_ && j < NCP_) {
        float dx = sTp[i][0] - sTp[j][0];
        float dy = sTp[i][1] - sTp[j][1];
        float dz = sTp[i][2] - sTp[j][2];
        val = sqrtf(fmaxf(dx*dx + dy*dy + dz*dz, 1e-10f));
      } else if (i < NCP_) {
        int c = j - NCP_;
        val = (c < 3) ? sTp[i][c] : 1.f;
      } else if (j < NCP_) {
        int c = i - NCP_;
        val = (c < 3) ? sTp[j][c] : 1.f;
      } else {
        val = 0.f;
      }
    } else {
      int c = j - NSYS;  // tv = flip(cp)*2-1 in rows < NCP, zeros below
      val = (i < NCP_) ? (cp[i*3 + (2 - c)] * 2.f - 1.f) : 0.f;
    }
    M[i*LDA + j] = val;
  }
  __syncthreads();

  // --- Gaussian elimination with partial pivoting ---
  for (int k = 0; k < NSYS; ++k) {
    if (tid == 0) {
      int p = k;
      float best = fabsf(M[k*LDA + k]);
      for (int r = k + 1; r < NSYS; ++r) {
        float v = fabsf(M[r*LDA + k]);
        if (v > best) { best = v; p = r; }
      }
      sPiv = p;
    }
    __syncthreads();
    int p = sPiv;
    if (p != k) {
      for (int c = k + tid; c < NCOLS; c += 256) {
        float a = M[k*LDA + c];
        float b = M[p*LDA + c];
        M[k*LDA + c] = b;
        M[p*LDA + c] = a;
      }
    }
    __syncthreads();
    float inv = 1.f / M[k*LDA + k];
    for (int r = k + 1 + tid; r < NSYS; r += 256) {
      float f = M[r*LDA + k] * inv;
      for (int c = k; c < NCOLS; ++c)
        M[r*LDA + c] -= f * M[k*LDA + c];
    }
    __syncthreads();
  }

  // --- back substitution (one thread per RHS column) ---
  if (tid < 3) {
    int c = NSYS + tid;
    for (int k = NSYS - 1; k >= 0; --k) {
      float a = M[k*LDA + c];
      for (int j = k + 1; j < NSYS; ++j) a -= M[k*LDA + j] * M[j*LDA + c];
      M[k*LDA + c] = a / M[k*LDA + k];
    }
  }
  __syncthreads();

  // --- pack solution ---
  if (tid < NCP_) {
    for (int c = 0; c < 3; ++c) ws[P_W + tid*4 + c] = M[tid*LDA + NSYS + c];
    ws[P_W + tid*4 + 3] = 0.f;
  }
  if (tid < 4) {
    for (int c = 0; c < 3; ++c) ws[P_V + tid*4 + c] = M[(NCP_ + tid)*LDA + NSYS + c];
    ws[P_V + tid*4 + 3] = 0.f;
  }
}

// ---------------------------------------------------------------------------
// Kernel 1: WMMA spline evaluation + trilinear sampling
// 128 threads (4 waves) per block, 1024 blocks, 32 queries per wave-tile.
// ---------------------------------------------------------------------------
__global__ __launch_bounds__(128) void warp_sample_kernel(
    const float* __restrict__ grid,
    const float* __restrict__ vol,
    const float* __restrict__ ws,
    float* __restrict__ out)
{
  __shared__ float sh_bmat[NCP_][4];       // 2 KB
  __shared__ float sh_wpad[NCP_][16];      // 8 KB  (w padded to N=16)
  __shared__ float sh_v[16];               // affine rows (4x4)
  __shared__ float sh_Rm[12];
  __shared__ float sh_toff[S_][4];
  __shared__ float sh_dq[4][16][NCP_];     // 32 KB (per-wave 16q x 128cp)
  __shared__ float sh_bg[4][32][4];        // 2 KB  (x,y,z,1.0) -> branchless A frag
  __shared__ float sh_n2[4][32];           // 0.5 KB (|q|^2 for C frag)
  __shared__ float sh_s[4][32][4];         // 2 KB  (sample locs)

  const int tid  = threadIdx.x;
  const int lane = tid & 31;
  const int wid  = tid >> 5;
  const int hf   = lane >> 4;   // 0/1 half-wave
  const int l16  = lane & 15;

  for (int e = tid; e < NCP_ * 4; e += 128) sh_bmat[e >> 2][e & 3] = ws[P_BMAT + e];
  for (int e = tid; e < NCP_ * 16; e += 128) {
    int j = e >> 4, n = e & 15;
    sh_wpad[j][n] = (n < 3) ? ws[P_W + j*4 + n] : 0.f;
  }
  if (tid < 16) sh_v[tid] = ws[P_V + tid];
  if (tid < 12) sh_Rm[tid] = (tid < 9) ? ws[P_RM + tid] : 0.f;
  if (tid < S_ * 4) sh_toff[tid >> 2][tid & 3] = ws[P_TOFF + tid];
  __syncthreads();

  const int nwaves = gridDim.x * 4;
  for (int T = blockIdx.x * 4 + wid; T < Q_ / 32; T += nwaves) {
    const int q0 = T * 32;
    const int q  = q0 + lane;
    const int s  = q >> 14;  // slice = q / (SZ*SZ)

    // rotated + offset query position (row-vector * Rm)
    float gx = grid[q*3 + 0], gy = grid[q*3 + 1], gz = grid[q*3 + 2];
    float bx = gx*sh_Rm[0] + gy*sh_Rm[3] + gz*sh_Rm[6] + sh_toff[s][0];
    float by = gx*sh_Rm[1] + gy*sh_Rm[4] + gz*sh_Rm[7] + sh_toff[s][1];
    float bz = gx*sh_Rm[2] + gy*sh_Rm[5] + gz*sh_Rm[8] + sh_toff[s][2];
    sh_bg[wid][lane][0] = bx;
    sh_bg[wid][lane][1] = by;
    sh_bg[wid][lane][2] = bz;
    sh_bg[wid][lane][3] = 1.0f;                    // homogeneous coord -> A frag K=3
    sh_n2[wid][lane] = bx*bx + by*by + bz*bz;      // |q|^2 -> C frag
    asm volatile("s_wait_dscnt 0" ::: "memory");   // intra-wave LDS fence

#pragma unroll
    for (int t = 0; t < 2; ++t) {
      const int mrow = t*16 + l16;

      // A fragment: rows = queries, K = (x,y,z,1); branchless ds_load_b64
      v2f a1;
      a1.x = sh_bg[wid][mrow][hf*2 + 0];
      a1.y = sh_bg[wid][mrow][hf*2 + 1];

      // C fragment: |q|^2 broadcast along rows
      v8f c1;
#pragma unroll
      for (int r = 0; r < 8; ++r) c1[r] = sh_n2[wid][t*16 + r + 8*hf];

      // Stage 1: 8 WMMAs -> squared distances for all 128 control points
#pragma unroll
      for (int cpb = 0; cpb < 8; ++cpb) {
        v2f b1;
        b1.x = sh_bmat[cpb*16 + l16][hf*2 + 0];
        b1.y = sh_bmat[cpb*16 + l16][hf*2 + 1];
        v8f d = __builtin_amdgcn_wmma_f32_16x16x4_f32(
            false, a1, false, b1, (short)0, c1, false, false);
#pragma unroll
        for (int r = 0; r < 8; ++r) {
          // single v_sqrt_f32 (no IEEE refinement sequence)
          float dq = __builtin_amdgcn_sqrtf(fmaxf(d[r], 1e-10f));
          sh_dq[wid][r + 8*hf][cpb*16 + l16] = dq;  // [query][cp]
        }
      }
      asm volatile("s_wait_dscnt 0" ::: "memory");

      // Stage 2: 32 chained WMMAs accumulate dq @ w (N padded to 16)
      v8f acc = {0.f, 0.f, 0.f, 0.f, 0.f, 0.f, 0.f, 0.f};
#pragma unroll
      for (int kc = 0; kc < 32; ++kc) {
        const int k0 = kc*4 + hf*2;
        v2f a2, b2;
        a2.x = sh_dq[wid][l16][k0 + 0];
        a2.y = sh_dq[wid][l16][k0 + 1];
        b2.x = sh_wpad[k0 + 0][l16];
        b2.y = sh_wpad[k0 + 1][l16];
        acc = __builtin_amdgcn_wmma_f32_16x16x4_f32(
            false, a2, false, b2, (short)0, acc, false, false);
      }

      // Epilogue: add affine part, scatter sample coords to LDS
      if (l16 < 3) {
        const int n = l16;
#pragma unroll
        for (int r = 0; r < 8; ++r) {
          const int m = t*16 + r + 8*hf;
          float val = acc[r]
                    + sh_bg[wid][m][0] * sh_v[0*4 + n]
                    + sh_bg[wid][m][1] * sh_v[1*4 + n]
                    + sh_bg[wid][m][2] * sh_v[2*4 + n]
                    + sh_v[3*4 + n];
          sh_s[wid][m][n] = val;
        }
      }
    }
    asm volatile("s_wait_dscnt 0" ::: "memory");

    // Trilinear sample: one query per lane
    {
      float x = sh_s[wid][lane][0];
      float y = sh_s[wid][lane][1];
      float z = sh_s[wid][lane][2];
      float fx = (x + 1.f) * 0.5f * (float)(SZ_ - 1);
      float fy = (y + 1.f) * 0.5f * (float)(SZ_ - 1);
      float fz = (z + 1.f) * 0.5f * (float)(SZ_ - 1);
      float x0 = floorf(fx), y0 = floorf(fy), z0 = floorf(fz);
      float tx = fx - x0, ty = fy - y0, tz = fz - z0;
      int ix0 = (int)x0, iy0 = (int)y0, iz0 = (int)z0;
      float outv = 0.f;
#pragma unroll
      for (int dz = 0; dz < 2; ++dz)
#pragma unroll
        for (int dy = 0; dy < 2; ++dy)
#pragma unroll
          for (int dx = 0; dx < 2; ++dx) {
            int xi = ix0 + dx, yi = iy0 + dy, zi = iz0 + dz;
            float wx = dx ? tx : 1.f - tx;
            float wy = dy ? ty : 1.f - ty;
            float wz = dz ? tz : 1.f - tz;
            bool ok = (xi >= 0) & (xi < SZ_) & (yi >= 0) & (yi < SZ_) &
                      (zi >= 0) & (zi < SZ_);
            int xc = xi < 0 ? 0 : (xi > SZ_ - 1 ? SZ_ - 1 : xi);
            int yc = yi < 0 ? 0 : (yi > SZ_ - 1 ? SZ_ - 1 : yi);
            int zc = zi < 0 ? 0 : (zi > SZ_ - 1 ? SZ_ - 1 : zi);
            float v = vol[(zc * SZ_ + yc) * SZ_ + xc];
            outv += v * (wx * wy * wz) * (ok ? 1.f : 0.f);
          }
      const int rem = q & (SZ_ * SZ_ - 1);
      const int i0 = rem >> 7;
      const int i1 = rem & (SZ_ - 1);
      out[(i0 * SZ_ + i1) * S_ + s] = outv;  // (1,1,128,128,16)
    }
  }
}

// ---------------------------------------------------------------------------
extern "C" void kernel_launch(void* const* d_in, const int* in_sizes, int n_in,
                              void* d_out, int out_size, void* d_ws, size_t ws_size,
                              hipStream_t stream) {
  (void)in_sizes; (void)n_in; (void)out_size; (void)ws_size;
  const float* wcp   = (const float*)d_in[0];  // warped_control_points
  const float* vol   = (const float*)d_in[1];  // vol
  const float* goff  = (const float*)d_in[2];  // global_offsets
  const float* pso   = (const float*)d_in[3];  // per_slice_offsets
  const float* grot  = (const float*)d_in[4];  // global_rotations
  const float* grid  = (const float*)d_in[5];  // grid
  const float* cp    = (const float*)d_in[6];  // control_points
  const float* coord = (const float*)d_in[7];  // coordinate_system
  float* out = (float*)d_out;
  float* ws  = (float*)d_ws;

  setup_kernel<<<dim3(1), dim3(256), 0, stream>>>(wcp, goff, pso, grot, cp, coord, ws);
  warp_sample_kernel<<<dim3(1024), dim3(128), 0, stream>>>(grid, vol, ws, out);
}